// BERTNet_75256416961146
// MI455X (gfx1250) — compile-verified
//
#include <hip/hip_runtime.h>
#include <math.h>

// Problem dimensions (match reference)
#define B_   64
#define S_   512
#define T_   256
#define C_   1024
#define D_   768
#define MLP_ 600
#define CLS_ 4

#define NACC 8                    // 16x(16*NACC) C footprint per wave
#define BCOLS (16 * NACC)         // 128 columns staged per block
#define BSTRIDE 68                // padded K-stride (dwords): bank = (4*col + k) % 64

typedef __attribute__((ext_vector_type(2))) float v2f;
typedef __attribute__((ext_vector_type(8))) float v8f;

// ---------------------------------------------------------------------------
// Kernel 1: segment mean. One block per (sentence, token); binary-search the
// contiguous subtoken run (seg_ids sorted per sentence), average over it.
// ---------------------------------------------------------------------------
__global__ __launch_bounds__(128) void segmean_kernel(
    const float* __restrict__ h,      // [B,S,D]
    const int*   __restrict__ seg,    // [B,S]
    float*       __restrict__ means)  // [B,T,D]
{
    const int bt = blockIdx.x;
    const int b = bt / T_;
    const int t = bt % T_;
    const int* sb = seg + (size_t)b * S_;

    int lo = 0, hi = S_;
    while (lo < hi) { int mid = (lo + hi) >> 1; if (sb[mid] < t) lo = mid + 1; else hi = mid; }
    int lo2 = lo, hi2 = S_;
    while (lo2 < hi2) { int mid = (lo2 + hi2) >> 1; if (sb[mid] < t + 1) lo2 = mid + 1; else hi2 = mid; }

    const int cnt = lo2 - lo;
    const float inv = (cnt > 0) ? (1.0f / (float)cnt) : 0.0f;

    const float* hb = h + (size_t)b * S_ * D_;
    float* mout = means + ((size_t)b * T_ + t) * D_;
    for (int d = threadIdx.x; d < D_; d += blockDim.x) {
        float s = 0.0f;
        for (int si = lo; si < lo2; ++si) s += hb[(size_t)si * D_ + d];
        mout[d] = s * inv;
    }
}

// ---------------------------------------------------------------------------
// Kernel 2: fp32 WMMA GEMM  C[Mr,N] = A[Mr,K] @ Bm[K,N]
// Block = 4 waves, each wave owns a 16 x 128 C tile (NACC=8 accumulators);
// one A fragment feeds 8 WMMAs. B chunk staged K-contiguous per column in
// LDS (stride 68 dwords -> conflict-free b64 fragment reads).
// Mr multiple of 64 (16384 here); K multiple of 64 (768 here).
// ---------------------------------------------------------------------------
__global__ __launch_bounds__(128) void gemm_wmma_f32(
    const float* __restrict__ A,
    const float* __restrict__ Bm,
    float*       __restrict__ Cm,
    int Mr, int N, int K)
{
    __shared__ float Bt[BCOLS * BSTRIDE];    // 128 cols x 64 K (padded) = 34 KB

    const int lane  = threadIdx.x & 31;
    const int wave  = threadIdx.x >> 5;
    const int nbase = blockIdx.x * BCOLS;
    const int mbase = (blockIdx.y * 4 + wave) * 16;
    const int half  = lane >> 4;    // 0: lanes 0-15 (K0,K1), 1: lanes 16-31 (K2,K3)
    const int lcol  = lane & 15;

    // A fragment rows: lanes 0-15 -> M=lane; lanes 16-31 -> M=lane-16
    const float* arow = A + (size_t)(mbase + lcol) * K + half * 2;

    // Each thread stages one B column (coalesced: per-k row is 512B contiguous)
    const int scol = threadIdx.x;            // 0..127
    const int gcol = nbase + scol;
    float* btcol = Bt + scol * BSTRIDE;

    v8f acc[NACC] = {};

    for (int kc = 0; kc < K; kc += 64) {
        if (gcol < N) {
            #pragma unroll 8
            for (int kk = 0; kk < 64; ++kk)
                btcol[kk] = Bm[(size_t)(kc + kk) * N + gcol];
        } else {
            #pragma unroll 8
            for (int kk = 0; kk < 64; ++kk)
                btcol[kk] = 0.0f;
        }
        __syncthreads();

        // A double-buffer across k-steps
        v2f af = *(const v2f*)(arow + kc);
        #pragma unroll
        for (int k0 = 0; k0 < 64; k0 += 4) {
            const int kb = k0 + half * 2;    // even -> 8B-aligned ds_load_b64
            // Batch-load all 8 B fragments (one ds_load_b64 each)
            v2f bf[NACC];
            #pragma unroll
            for (int t = 0; t < NACC; ++t)
                bf[t] = *(const v2f*)(Bt + (lcol + t * 16) * BSTRIDE + kb);
            // Prefetch next A fragment behind the WMMA burst
            v2f af_next = af;
            if (k0 + 4 < 64)
                af_next = *(const v2f*)(arow + kc + k0 + 4);
            #pragma unroll
            for (int t = 0; t < NACC; ++t)
                acc[t] = __builtin_amdgcn_wmma_f32_16x16x4_f32(
                    /*neg_a=*/false, af, /*neg_b=*/false, bf[t],
                    /*c_mod=*/(short)0, acc[t], /*reuse_a=*/false, /*reuse_b=*/false);
            af = af_next;
        }
        __syncthreads();
    }

    // C layout: VGPR i -> M = i (lanes 0-15) / 8+i (lanes 16-31); N = lcol
    #pragma unroll
    for (int t = 0; t < NACC; ++t) {
        const int col = nbase + t * 16 + lcol;
        if (col < N) {
            #pragma unroll
            for (int i = 0; i < 8; ++i) {
                const int row = mbase + (half ? (8 + i) : i);
                Cm[(size_t)row * N + col] = acc[t][i];
            }
        }
    }
}

// ---------------------------------------------------------------------------
// Kernel 3: per-config head. One wave per config row:
//   hid = tanh(b1 + P[left] + Q[right])   (sentinel -> omit term)
//   logits = hid @ W2 + b2 ; softmax over 4 classes
// ---------------------------------------------------------------------------
__global__ __launch_bounds__(128) void head_kernel(
    const int*   __restrict__ conf,   // [B,C,2]
    const float* __restrict__ P,      // [B*T, MLP]
    const float* __restrict__ Q,      // [B*T, MLP]
    const float* __restrict__ b1,     // [MLP]
    const float* __restrict__ W2,     // [MLP, CLS]
    const float* __restrict__ b2,     // [CLS]
    float*       __restrict__ out)    // [B*C, CLS]
{
    const int lane = threadIdx.x & 31;
    const int wave = threadIdx.x >> 5;
    const int r = blockIdx.x * 4 + wave;      // config row in [0, B*C)
    const int b = r / C_;

    const int left  = conf[2 * r + 0] - 1;
    const int right = conf[2 * r + 1] - 1;
    const float* pr = (left  >= 0) ? (P + ((size_t)b * T_ + left ) * MLP_) : nullptr;
    const float* qr = (right >= 0) ? (Q + ((size_t)b * T_ + right) * MLP_) : nullptr;

    float lg0 = 0.f, lg1 = 0.f, lg2 = 0.f, lg3 = 0.f;
    for (int j = lane; j < MLP_; j += 32) {
        float x = b1[j];
        if (pr) x += pr[j];
        if (qr) x += qr[j];
        const float hd = tanhf(x);
        const float4 w = *(const float4*)(W2 + 4 * j);  // row j of W2, 16B aligned
        lg0 += hd * w.x; lg1 += hd * w.y; lg2 += hd * w.z; lg3 += hd * w.w;
    }

    #pragma unroll
    for (int off = 16; off >= 1; off >>= 1) {
        lg0 += __shfl_xor(lg0, off, 32);
        lg1 += __shfl_xor(lg1, off, 32);
        lg2 += __shfl_xor(lg2, off, 32);
        lg3 += __shfl_xor(lg3, off, 32);
    }

    if (lane == 0) {
        lg0 += b2[0]; lg1 += b2[1]; lg2 += b2[2]; lg3 += b2[3];
        const float m  = fmaxf(fmaxf(lg0, lg1), fmaxf(lg2, lg3));
        const float e0 = expf(lg0 - m), e1 = expf(lg1 - m);
        const float e2 = expf(lg2 - m), e3 = expf(lg3 - m);
        const float inv = 1.0f / (e0 + e1 + e2 + e3);
        out[4 * r + 0] = e0 * inv;
        out[4 * r + 1] = e1 * inv;
        out[4 * r + 2] = e2 * inv;
        out[4 * r + 3] = e3 * inv;
    }
}

// ---------------------------------------------------------------------------
extern "C" void kernel_launch(void* const* d_in, const int* in_sizes, int n_in,
                              void* d_out, int out_size, void* d_ws, size_t ws_size,
                              hipStream_t stream) {
    const float* h    = (const float*)d_in[0];  // [B,S,D]
    const int*   seg  = (const int*)  d_in[1];  // [B,S]
    const int*   conf = (const int*)  d_in[2];  // [B,C,2]
    const float* W1   = (const float*)d_in[3];  // [2D,MLP]
    const float* b1   = (const float*)d_in[4];  // [MLP]
    const float* W2   = (const float*)d_in[5];  // [MLP,CLS]
    const float* b2   = (const float*)d_in[6];  // [CLS]
    float* out = (float*)d_out;                 // [B*C, CLS]

    // Workspace layout (fp32): means | P | Q
    float* means = (float*)d_ws;                              // B*T*D
    float* P     = means + (size_t)B_ * T_ * D_;              // B*T*MLP
    float* Q     = P     + (size_t)B_ * T_ * MLP_;            // B*T*MLP

    // 1) Per-token segment means
    segmean_kernel<<<B_ * T_, 128, 0, stream>>>(h, seg, means);

    // 2) P = means @ W1[0:D], Q = means @ W1[D:2D]  (fp32 WMMA GEMMs)
    dim3 g2((MLP_ + BCOLS - 1) / BCOLS, (B_ * T_) / 64);      // 5 x 256
    gemm_wmma_f32<<<g2, 128, 0, stream>>>(means, W1,                      P, B_ * T_, MLP_, D_);
    gemm_wmma_f32<<<g2, 128, 0, stream>>>(means, W1 + (size_t)D_ * MLP_,  Q, B_ * T_, MLP_, D_);

    // 3) Per-config head: gather + tanh + 600x4 GEMV + softmax
    head_kernel<<<(B_ * C_) / 4, 128, 0, stream>>>(conf, P, Q, b1, W2, b2, out);
}